// SOM_49228915147270
// MI455X (gfx1250) — compile-verified
//
#include <hip/hip_runtime.h>
#include <cmath>

// SOM dimensions (fixed by the reference)
constexpr int Hh = 64, Ww = 64, Dd = 256, Bb = 8192, Nn = 4096, ITERS = 10;

typedef float v2f __attribute__((ext_vector_type(2)));
typedef float v8f __attribute__((ext_vector_type(8)));
typedef int   v4i __attribute__((ext_vector_type(4)));

#define XLD 260   // padded LDS row stride (floats) for the staged input strips
#define EB  128   // batch chunk for the update GEMM
#define ELD 132   // padded LDS row stride (floats) for the eff chunk

// gfx1250 async global->LDS copy path (ASYNCcnt), with safe fallback.
#if defined(__has_builtin)
#if __has_builtin(__builtin_amdgcn_global_load_async_to_lds_b128) && \
    __has_builtin(__builtin_amdgcn_s_wait_asynccnt)
#define HAVE_ASYNC_LDS 1
#endif
#endif

#if defined(HAVE_ASYNC_LDS)
typedef __attribute__((address_space(1))) v4i gbl_v4i;   // global (AS1)
typedef __attribute__((address_space(3))) v4i lds_v4i;   // LDS (AS3)
#endif

// ---------------------------------------------------------------------------
// w2[n] = sum_d w[n][d]^2   (one wave per row, shuffle reduce)
// ---------------------------------------------------------------------------
__global__ __launch_bounds__(256)
void som_w2(const float* __restrict__ w, float* __restrict__ w2) {
  const int lane = threadIdx.x & 31;
  const int wv   = threadIdx.x >> 5;
  const int n    = blockIdx.x * 8 + wv;
  const float* row = &w[(size_t)n * Dd];
  float s = 0.f;
#pragma unroll
  for (int j = 0; j < Dd; j += 32) {
    float v = row[j + lane];
    s += v * v;
  }
#pragma unroll
  for (int off = 16; off >= 1; off >>= 1) s += __shfl_xor(s, off, 32);
  if (lane == 0) w2[n] = s;
}

// ---------------------------------------------------------------------------
// 32 batch rows per workgroup (two 16-row A strips in LDS). Each codebook
// B-fragment (global) feeds TWO v_wmma_f32_16x16x4_f32 ops -> halves the L2
// traffic on w (4 MB) and doubles WMMA per vmem load.
// argmin_n of dist2 = |x|^2 - 2 x.w + |w|^2 needs only (w2[n] - 2*x.w).
// ---------------------------------------------------------------------------
__global__ __launch_bounds__(256)
void som_dist_argmin(const float* __restrict__ x, const float* __restrict__ w,
                     const float* __restrict__ w2, int* __restrict__ bmu) {
  __shared__ float xs[32 * XLD];
  __shared__ float pmv[8][32];
  __shared__ int   pmi[8][32];

  const int tid  = threadIdx.x;
  const int lane = tid & 31;
  const int wv   = tid >> 5;
  const int b0   = blockIdx.x * 32;

  // Stage 32 input rows (32 x 256 f32) into LDS.
#if defined(HAVE_ASYNC_LDS)
  // Async DMA path: 16B per lane per op, tracked by ASYNCcnt.
  for (int t = tid; t < 32 * (Dd / 4); t += 256) {
    const int r = t >> 6;            // row 0..31 (64 x 16B groups per row)
    const int c = (t & 63) << 2;     // float column, multiple of 4 -> 16B aligned
    __builtin_amdgcn_global_load_async_to_lds_b128(
        (gbl_v4i*)&x[(size_t)(b0 + r) * Dd + c],
        (lds_v4i*)&xs[r * XLD + c], 0, 0);
  }
  __builtin_amdgcn_s_wait_asynccnt(0);
  __syncthreads();
#else
  for (int t = tid; t < 32 * Dd; t += 256) {
    const int r = t >> 8;      // /256
    const int c = t & 255;
    xs[r * XLD + c] = x[(size_t)(b0 + r) * Dd + c];
  }
  __syncthreads();
#endif

  const int m  = lane & 15;        // A-row (M) for lanes; also C/B column (N%16)
  const int kb = (lane >> 4) * 2;  // K sub-offset: lanes 0-15 -> K{0,1}, 16-31 -> K{2,3}

  float best[2][8];
  int   bidx[2][8];
#pragma unroll
  for (int s = 0; s < 2; ++s)
#pragma unroll
    for (int r = 0; r < 8; ++r) { best[s][r] = 3.4e38f; bidx[s][r] = 0; }

  // 256 column tiles of 16, interleaved over 8 waves -> 32 tiles/wave.
  for (int t = 0; t < 32; ++t) {
    const int n0c = (wv + (t << 3)) << 4;
    if (t < 31) {
      // hint the next tile's codebook rows toward the caches (global_prefetch_b8)
      const int npf = (wv + ((t + 1) << 3)) << 4;
      __builtin_prefetch(&w[(size_t)(npf + m) * Dd], 0, 3);
    }
    v8f acc0 = {}, acc1 = {};
#pragma unroll 8
    for (int k = 0; k < Dd; k += 4) {
      // B fragment (4x16 f32) = w^T: B[kr][nc] = w[n0c+nc][kr]; shared by both strips.
      v2f bf = *(const v2f*)&w[(size_t)(n0c + m) * Dd + k + kb];
      // A fragments (16x4 f32) for batch strips 0 and 1.
      v2f a0 = *(const v2f*)&xs[m * XLD + k + kb];
      v2f a1 = *(const v2f*)&xs[(16 + m) * XLD + k + kb];
      acc0 = __builtin_amdgcn_wmma_f32_16x16x4_f32(false, a0, false, bf,
                                                   (short)0, acc0, false, false);
      acc1 = __builtin_amdgcn_wmma_f32_16x16x4_f32(false, a1, false, bf,
                                                   (short)0, acc1, false, false);
    }
    // C slot (vgpr r, lane): row = r + 8*(lane>=16), col = lane&15 == m
    const float w2n = w2[n0c + m];
    const int   n   = n0c + m;
#pragma unroll
    for (int r = 0; r < 8; ++r) {
      const float v0 = w2n - 2.0f * acc0[r];
      const float v1 = w2n - 2.0f * acc1[r];
      if (v0 < best[0][r]) { best[0][r] = v0; bidx[0][r] = n; }
      if (v1 < best[1][r]) { best[1][r] = v1; bidx[1][r] = n; }
    }
  }

  // Reduce the 16 column-candidates of each row across each 16-lane half.
#pragma unroll
  for (int s = 0; s < 2; ++s) {
#pragma unroll
    for (int r = 0; r < 8; ++r) {
      float v  = best[s][r];
      int   ix = bidx[s][r];
#pragma unroll
      for (int off = 8; off >= 1; off >>= 1) {
        const float ov = __shfl_xor(v, off, 16);
        const int   oi = __shfl_xor(ix, off, 16);
        if (ov < v || (ov == v && oi < ix)) { v = ov; ix = oi; }
      }
      if (m == 0) {
        const int row = s * 16 + r + ((lane >> 4) << 3);
        pmv[wv][row] = v;
        pmi[wv][row] = ix;
      }
    }
  }
  __syncthreads();

  // Combine the 8 waves' partials; smaller index wins ties (jnp.argmin order).
  if (tid < 32) {
    float v  = pmv[0][tid];
    int   ix = pmi[0][tid];
    for (int g = 1; g < 8; ++g) {
      const float ov = pmv[g][tid];
      const int   oi = pmi[g][tid];
      if (ov < v || (ov == v && oi < ix)) { v = ov; ix = oi; }
    }
    bmu[b0 + tid] = ix;
  }
}

// ---------------------------------------------------------------------------
// One workgroup per 32-node strip. Streams B in chunks of 128:
//   eff[b,n] = lr * exp(-0.5*d2/sigma2) * (d2 < sigma2)   (d2 = lattice dist^2)
// Each inputs B-fragment (global) feeds TWO eff A-strips -> 4 WMMAs per pair
// of global loads. Then w += (acc - w*sum_b eff)/B, fully element-owned.
// ---------------------------------------------------------------------------
__global__ __launch_bounds__(256)
void som_update(const float* __restrict__ x, float* __restrict__ w,
                const int* __restrict__ bmu, float lr, float sigma2) {
  __shared__ float eff[32 * ELD];   // [nn][bb], padded
  __shared__ float sp[256];
  __shared__ float srow[32];

  const int tid  = threadIdx.x;
  const int lane = tid & 31;
  const int wv   = tid >> 5;
  const int n0   = blockIdx.x * 32;

  const int m  = lane & 15;
  const int kb = (lane >> 4) * 2;

  const int nn = tid & 31;        // eff row this thread fills (0..31)
  const int bg = tid >> 5;        // 0..7 batch-column group
  const int ni = (n0 + nn) >> 6;
  const int nj = (n0 + nn) & 63;

  const int d0 = (wv << 4);
  const int d1 = ((wv + 8) << 4);

  v8f acc00 = {}, acc01 = {}, acc10 = {}, acc11 = {};
  float slocal = 0.f;

  for (int cb = 0; cb < Bb; cb += EB) {
    // Build the eff chunk (32 n-rows x 128 b-cols) in LDS.
#pragma unroll
    for (int j = 0; j < EB / 8; ++j) {
      const int bb  = bg + (j << 3);
      const int bmv = bmu[cb + bb];
      const float di = (float)(ni - (bmv >> 6));
      const float dj = (float)(nj - (bmv & 63));
      const float d2 = di * di + dj * dj;
      float e = 0.f;
      if (d2 < sigma2) e = lr * __expf(-0.5f * d2 / sigma2);
      eff[nn * ELD + bb] = e;
      slocal += e;
    }
    __syncthreads();

    // WMMA: A = eff strips (two 16-row strips), B = inputs rows for two d tiles.
#pragma unroll 2
    for (int k = 0; k < EB; k += 4) {
      v2f a0 = *(const v2f*)&eff[m * ELD + k + kb];
      v2f a1 = *(const v2f*)&eff[(16 + m) * ELD + k + kb];
      const size_t krow = (size_t)(cb + k + kb);
      v2f bf0, bf1;
      bf0.x = x[krow * Dd + d0 + m];
      bf0.y = x[(krow + 1) * Dd + d0 + m];
      bf1.x = x[krow * Dd + d1 + m];
      bf1.y = x[(krow + 1) * Dd + d1 + m];
      acc00 = __builtin_amdgcn_wmma_f32_16x16x4_f32(false, a0, false, bf0,
                                                    (short)0, acc00, false, false);
      acc01 = __builtin_amdgcn_wmma_f32_16x16x4_f32(false, a0, false, bf1,
                                                    (short)0, acc01, false, false);
      acc10 = __builtin_amdgcn_wmma_f32_16x16x4_f32(false, a1, false, bf0,
                                                    (short)0, acc10, false, false);
      acc11 = __builtin_amdgcn_wmma_f32_16x16x4_f32(false, a1, false, bf1,
                                                    (short)0, acc11, false, false);
    }
    __syncthreads();
  }

  // Per-n sums of eff across the batch (8 partials per row).
  sp[tid] = slocal;
  __syncthreads();
  if (tid < 32) {
    float s = 0.f;
    for (int g = 0; g < 8; ++g) s += sp[(g << 5) + tid];
    srow[tid] = s;
  }
  __syncthreads();

  // w += (acc - w * s) / B ; each (n,d) owned by exactly one slot.
  const int rbase = (lane >> 4) * 8;
#pragma unroll
  for (int r = 0; r < 8; ++r) {
    const int row0 = r + rbase;          // strip 0 row
    const int row1 = 16 + r + rbase;     // strip 1 row
    const float s0 = srow[row0];
    const float s1 = srow[row1];
    const size_t i00 = (size_t)(n0 + row0) * Dd + d0 + m;
    const size_t i01 = (size_t)(n0 + row0) * Dd + d1 + m;
    const size_t i10 = (size_t)(n0 + row1) * Dd + d0 + m;
    const size_t i11 = (size_t)(n0 + row1) * Dd + d1 + m;
    const float w00 = w[i00], w01 = w[i01], w10 = w[i10], w11 = w[i11];
    w[i00] = w00 + (acc00[r] - w00 * s0) * (1.0f / 8192.0f);
    w[i01] = w01 + (acc01[r] - w01 * s0) * (1.0f / 8192.0f);
    w[i10] = w10 + (acc10[r] - w10 * s1) * (1.0f / 8192.0f);
    w[i11] = w11 + (acc11[r] - w11 * s1) * (1.0f / 8192.0f);
  }
}

// ---------------------------------------------------------------------------
// out[b,:] = w[bmu[b],:]
// ---------------------------------------------------------------------------
__global__ __launch_bounds__(256)
void som_gather(const float* __restrict__ w, const int* __restrict__ bmu,
                float* __restrict__ out) {
  const int b = blockIdx.x;
  const int d = threadIdx.x;
  out[(size_t)b * Dd + d] = w[(size_t)bmu[b] * Dd + d];
}

// ---------------------------------------------------------------------------
extern "C" void kernel_launch(void* const* d_in, const int* in_sizes, int n_in,
                              void* d_out, int out_size, void* d_ws, size_t ws_size,
                              hipStream_t stream) {
  const float* x      = (const float*)d_in[0];   // [B, D]
  const float* w_init = (const float*)d_in[1];   // [N, D]
  float* out = (float*)d_out;                    // [B, D]

  float* w   = (float*)d_ws;                     // working codebook, N*D
  float* w2  = w + (size_t)Nn * Dd;              // N
  int*   bmu = (int*)(w2 + Nn);                  // B

  (void)hipMemcpyAsync(w, w_init, (size_t)Nn * Dd * sizeof(float),
                       hipMemcpyDeviceToDevice, stream);

  const double TIME_CONST = 10.0 / log(32.0);    // NUM_ITERS / log(RADIUS)
  for (int i = 0; i < ITERS; ++i) {
    const float lr  = 0.1f * expf(-(float)i / 10.0f);
    const float sig = 32.0f * (float)exp(-(double)i / TIME_CONST);
    const float sigma2 = sig * sig;

    som_w2<<<Nn / 8, 256, 0, stream>>>(w, w2);
    som_dist_argmin<<<Bb / 32, 256, 0, stream>>>(x, w, w2, bmu);
    som_update<<<Nn / 32, 256, 0, stream>>>(x, w, bmu, lr, sigma2);
  }
  som_gather<<<Bb, 256, 0, stream>>>(w, bmu, out);
}